// PointBranch_7593502179721
// MI455X (gfx1250) — compile-verified
//
#include <hip/hip_runtime.h>
#include <hip/hip_bf16.h>

// ---------------------------------------------------------------------------
// PointBranch pipeline for MI455X (gfx1250, wave32).
// fp32 end-to-end; all GEMM-shaped work uses V_WMMA_F32_16X16X4_F32.
// Weights staged in LDS (shared across all waves of a WGP); no divergent
// loads inside WMMA streams.
// ---------------------------------------------------------------------------

typedef float v2f __attribute__((ext_vector_type(2)));
typedef float v8f __attribute__((ext_vector_type(8)));

__device__ __forceinline__ v8f wmma4(v2f a, v2f b, v8f c) {
  // D = A(16x4,f32) * B(4x16,f32) + C(16x16,f32)
  return __builtin_amdgcn_wmma_f32_16x16x4_f32(
      /*neg_a=*/false, a, /*neg_b=*/false, b,
      /*c_mod=*/(short)0, c, /*reuse_a=*/false, /*reuse_b=*/false);
}

#define N_PTS 16384
#define KNN   16
#define FEAT  64
#define MHID  128
#define MOUT  256

// ===========================================================================
// Kernel 0: zero-pad ew1 (6x64) -> ew1p (8x64) so EdgeConv B-frag loads are
// unconditional (no exec-divergent loads in the WMMA stream).
// ===========================================================================
__global__ __launch_bounds__(256) void pad_ew1_kernel(const float* __restrict__ ew1,
                                                      float* __restrict__ ew1p) {
  int i = blockIdx.x * 256 + threadIdx.x;
  if (i < 8 * FEAT) ew1p[i] = (i < 6 * FEAT) ? ew1[i] : 0.f;
}

// ===========================================================================
// Kernel 1: exact KNN (top-16) per point.  One wave per query point.
// Per-lane register top-16 (swap-chain insertion, static indexing only),
// then a 512-entry LDS merge with 16 rounds of wave-min extraction using
// 64-bit (distbits<<32 | pos) keys over wave32 shuffles.
// ===========================================================================
__global__ __launch_bounds__(256) void knn_kernel(const float* __restrict__ pts,
                                                  int* __restrict__ knn_idx) {
  __shared__ float s_d[8][512];
  __shared__ int   s_i[8][512];
  const int wave = threadIdx.x >> 5;
  const int lane = threadIdx.x & 31;
  const int q = blockIdx.x * 8 + wave;

  const float qx = pts[q * 3 + 0], qy = pts[q * 3 + 1], qz = pts[q * 3 + 2];

  float bd[KNN]; int bi[KNN];
#pragma unroll
  for (int t = 0; t < KNN; ++t) { bd[t] = __builtin_inff(); bi[t] = -1; }

  for (int j = lane; j < N_PTS; j += 32) {
    float dx = pts[j * 3 + 0] - qx;
    float dy = pts[j * 3 + 1] - qy;
    float dz = pts[j * 3 + 2] - qz;
    float d = dx * dx + dy * dy + dz * dz;
    if (j == q) d = __builtin_inff();   // diagonal excluded
    int jj = j;
    if (d < bd[KNN - 1]) {
#pragma unroll
      for (int t = 0; t < KNN; ++t) {
        if (d < bd[t]) {
          float td = bd[t]; int ti = bi[t];
          bd[t] = d; bi[t] = jj; d = td; jj = ti;
        }
      }
    }
  }

#pragma unroll
  for (int t = 0; t < KNN; ++t) {
    s_d[wave][lane * KNN + t] = bd[t];
    s_i[wave][lane * KNN + t] = bi[t];
  }
  __syncthreads();

  for (int k = 0; k < KNN; ++k) {
    float mind = __builtin_inff(); int minp = lane;
#pragma unroll
    for (int t = 0; t < KNN; ++t) {
      int p = t * 32 + lane;
      float dv = s_d[wave][p];
      if (dv < mind) { mind = dv; minp = p; }
    }
    unsigned long long key =
        (((unsigned long long)__float_as_uint(mind)) << 32) | (unsigned)minp;
#pragma unroll
    for (int off = 16; off >= 1; off >>= 1) {
      unsigned long long o = __shfl_xor(key, off, 32);
      if (o < key) key = o;
    }
    int wpos = (int)(key & 0xffffffffu);
    if (lane == (wpos & 31)) s_d[wave][wpos] = __builtin_inff();
    if (lane == 0) knn_idx[q * KNN + k] = s_i[wave][wpos];
    __syncthreads();
  }
}

// ===========================================================================
// Kernel 2: EdgeConv.  One wave per point; the point's 16 edges are the
// 16 M-rows of the WMMA tiles.  ew1p (8x64, 2 KB) and ew2 (64x64, 16 KB)
// are fully LDS-resident per block; all B-frags come from LDS.
// Stage1: ef(16x8 padded) @ ew1p + eb1, ReLU -> LDS.
// Stage2: h1(16x64) @ ew2 + eb2, then max over 16 edge rows -> feats.
// ===========================================================================
__global__ __launch_bounds__(128) void edgeconv_kernel(
    const float* __restrict__ pts, const int* __restrict__ knn_idx,
    const float* __restrict__ ew1p, const float* __restrict__ eb1,
    const float* __restrict__ ew2, const float* __restrict__ eb2,
    float* __restrict__ feats) {
  __shared__ float s_w1[8 * FEAT];        //  2 KB
  __shared__ float s_w2[FEAT * FEAT];     // 16 KB
  __shared__ float s_h1[4][16 * FEAT];    // 16 KB
  const int tid = threadIdx.x;
  const int wave = tid >> 5;
  const int lane = tid & 31;
  const int nl = lane & 15;              // M row / N col within tile
  const int kh = lane >> 4;              // K-half select for A/B frags
  const int q = blockIdx.x * 4 + wave;

  // cooperative weight staging (uniform, full block)
  for (int e = tid; e < 8 * FEAT; e += 128) s_w1[e] = ew1p[e];
  for (int e = tid; e < FEAT * FEAT; e += 128) s_w2[e] = ew2[e];

  const float pix = pts[q * 3 + 0], piy = pts[q * 3 + 1], piz = pts[q * 3 + 2];
  const int j = knn_idx[q * KNN + nl];   // lanes l and l+16 share row M=nl
  float e[8];
  e[0] = pix; e[1] = piy; e[2] = piz;
  e[3] = pts[j * 3 + 0] - pix;
  e[4] = pts[j * 3 + 1] - piy;
  e[5] = pts[j * 3 + 2] - piz;
  e[6] = 0.f; e[7] = 0.f;
  __syncthreads();

  // ---- stage 1: 4 N-tiles x 2 K-steps (K padded 6->8) ----
#pragma unroll
  for (int nt = 0; nt < 4; ++nt) {
    const int col = nt * 16 + nl;
    const float bias = eb1[col];
    v8f acc;
#pragma unroll
    for (int g = 0; g < 8; ++g) acc[g] = bias;
#pragma unroll
    for (int ks = 0; ks < 2; ++ks) {
      const int K0 = ks * 4 + kh * 2;
      v2f a;
      a.x = kh ? e[ks * 4 + 2] : e[ks * 4 + 0];
      a.y = kh ? e[ks * 4 + 3] : e[ks * 4 + 1];
      v2f b;
      b.x = s_w1[K0 * FEAT + col];
      b.y = s_w1[(K0 + 1) * FEAT + col];
      acc = wmma4(a, b, acc);
    }
#pragma unroll
    for (int g = 0; g < 8; ++g) {
      float v = acc[g];
      v = v > 0.f ? v : 0.f;                       // ReLU
      s_h1[wave][(g + 8 * kh) * FEAT + col] = v;   // C layout -> row-major LDS
    }
  }
  __syncthreads();

  // ---- stage 2: 4 N-tiles x 16 K-steps, then max over M rows ----
#pragma unroll
  for (int nt = 0; nt < 4; ++nt) {
    const int col = nt * 16 + nl;
    const float bias = eb2[col];
    v8f acc;
#pragma unroll
    for (int g = 0; g < 8; ++g) acc[g] = bias;
#pragma unroll
    for (int ks = 0; ks < 16; ++ks) {
      const int K0 = ks * 4 + kh * 2;
      v2f a;
      a.x = s_h1[wave][nl * FEAT + K0];
      a.y = s_h1[wave][nl * FEAT + K0 + 1];
      v2f b;
      b.x = s_w2[K0 * FEAT + col];
      b.y = s_w2[(K0 + 1) * FEAT + col];
      acc = wmma4(a, b, acc);
    }
    float m = acc[0];
#pragma unroll
    for (int g = 1; g < 8; ++g) m = fmaxf(m, acc[g]);
    m = fmaxf(m, __shfl_xor(m, 16, 32));           // combine both M halves
    if (lane < 16) feats[q * FEAT + col] = m;
  }
}

// ===========================================================================
// Kernel 3: farthest point sampling.  Single 1024-thread block; distance
// array lives in registers (<=16 elements / thread); each of the n_sample-1
// dependent steps does a shuffle+LDS argmax over the block.
// ===========================================================================
__global__ __launch_bounds__(1024) void fps_kernel(const float* __restrict__ src,
                                                   int n_src, int n_sample,
                                                   int* __restrict__ out_idx) {
  __shared__ unsigned long long s_wkey[32];
  __shared__ float s_cur[3];
  const int tid = threadIdx.x;
  const int lane = tid & 31;
  const int wave = tid >> 5;

  float px[16], py[16], pz[16], d[16];
  float c0 = src[0], c1 = src[1], c2 = src[2];
#pragma unroll
  for (int t = 0; t < 16; ++t) {
    int j = tid + t * 1024;
    if (j < n_src) {
      px[t] = src[j * 3 + 0]; py[t] = src[j * 3 + 1]; pz[t] = src[j * 3 + 2];
      float dx = px[t] - c0, dy = py[t] - c1, dz = pz[t] - c2;
      d[t] = dx * dx + dy * dy + dz * dz;
    } else {
      px[t] = 0.f; py[t] = 0.f; pz[t] = 0.f; d[t] = -1.0f;  // sentinel
    }
  }
  if (tid == 0) out_idx[0] = 0;

  for (int s = 1; s < n_sample; ++s) {
    float md = -1.0f; int mj = 0;
#pragma unroll
    for (int t = 0; t < 16; ++t) {
      if (d[t] > md) { md = d[t]; mj = tid + t * 1024; }
    }
    // max-key; invert index so ties resolve to the smallest index (argmax rule)
    unsigned long long key = (((unsigned long long)__float_as_uint(md)) << 32) |
                             (unsigned)(0xffffffffu - (unsigned)mj);
#pragma unroll
    for (int off = 16; off >= 1; off >>= 1) {
      unsigned long long o = __shfl_xor(key, off, 32);
      if (o > key) key = o;
    }
    if (lane == 0) s_wkey[wave] = key;
    __syncthreads();
    if (wave == 0) {
      unsigned long long k2 = s_wkey[lane];
#pragma unroll
      for (int off = 16; off >= 1; off >>= 1) {
        unsigned long long o = __shfl_xor(k2, off, 32);
        if (o > k2) k2 = o;
      }
      if (lane == 0) {
        int bj = (int)(0xffffffffu - (unsigned)(k2 & 0xffffffffu));
        out_idx[s] = bj;
        s_cur[0] = src[bj * 3 + 0];
        s_cur[1] = src[bj * 3 + 1];
        s_cur[2] = src[bj * 3 + 2];
      }
    }
    __syncthreads();
    c0 = s_cur[0]; c1 = s_cur[1]; c2 = s_cur[2];
#pragma unroll
    for (int t = 0; t < 16; ++t) {
      float dx = px[t] - c0, dy = py[t] - c1, dz = pz[t] - c2;
      float nd = dx * dx + dy * dy + dz * dz;
      d[t] = (d[t] < 0.f) ? d[t] : fminf(d[t], nd);
    }
  }
}

// ===========================================================================
// Kernel 4: row gather  dst[i][:] = src[idx[i]][:]
// ===========================================================================
__global__ __launch_bounds__(256) void gather_rows(const float* __restrict__ src,
                                                   const int* __restrict__ idx,
                                                   float* __restrict__ dst,
                                                   int rows, int width) {
  int i = blockIdx.x * 256 + threadIdx.x;
  int total = rows * width;
  if (i < total) {
    int r = i / width, c = i - r * width;
    dst[i] = src[idx[r] * width + c];
  }
}

// ===========================================================================
// Kernel 5: nearest-neighbor assignment + feature gather.  One wave per
// sampled point: argmin over all 16384 base points (min-key wave reduce,
// tie -> smallest index, matching jnp.argmin), then copy feats[nn].
// ===========================================================================
__global__ __launch_bounds__(256) void nn_feats_kernel(
    const float* __restrict__ sp, const float* __restrict__ pts,
    const float* __restrict__ feats, float* __restrict__ sf) {
  const int wave = threadIdx.x >> 5;
  const int lane = threadIdx.x & 31;
  const int s = blockIdx.x * 8 + wave;

  const float sx = sp[s * 3 + 0], sy = sp[s * 3 + 1], sz = sp[s * 3 + 2];
  float best = __builtin_inff(); int bj = 0;
  for (int j = lane; j < N_PTS; j += 32) {
    float dx = pts[j * 3 + 0] - sx;
    float dy = pts[j * 3 + 1] - sy;
    float dz = pts[j * 3 + 2] - sz;
    float dd = dx * dx + dy * dy + dz * dz;
    if (dd < best) { best = dd; bj = j; }
  }
  unsigned long long key =
      (((unsigned long long)__float_as_uint(best)) << 32) | (unsigned)bj;
#pragma unroll
  for (int off = 16; off >= 1; off >>= 1) {
    unsigned long long o = __shfl_xor(key, off, 32);
    if (o < key) key = o;
  }
  const int nn = (int)(key & 0xffffffffu);
#pragma unroll
  for (int c = 0; c < 2; ++c)
    sf[s * FEAT + c * 32 + lane] = feats[nn * FEAT + c * 32 + lane];
}

// ===========================================================================
// Kernel 6: scale MLP  lf = relu(sf @ w1 + b1) @ w2 + b2
// sf: rows x 64, w1: 64x128, w2: 128x256.  One wave per 16-row tile,
// 4 waves per block.  Per-N-tile weight slices staged in LDS (shared by all
// 4 waves); stage-1 A rows register-cached once per wave.  40 KB LDS/block.
// 640 v_wmma_f32_16x16x4_f32 per wave.
// ===========================================================================
__global__ __launch_bounds__(128) void mlp_kernel(
    const float* __restrict__ sf, const float* __restrict__ w1,
    const float* __restrict__ b1, const float* __restrict__ w2,
    const float* __restrict__ b2, float* __restrict__ out) {
  __shared__ float s_h[4][16 * MHID];    // 32 KB intermediate h
  __shared__ float s_b[MHID * 16];       //  8 KB B-tile staging
  const int tid = threadIdx.x;
  const int wave = tid >> 5;
  const int lane = tid & 31;
  const int nl = lane & 15;
  const int kh = lane >> 4;
  const int row0 = (blockIdx.x * 4 + wave) * 16;

  // Register-cache this wave's stage-1 A rows: lane needs cols {4k+2kh, +1}.
  float a0[16], a1[16];
#pragma unroll
  for (int ks = 0; ks < 16; ++ks) {
    const int K0 = ks * 4 + kh * 2;
    a0[ks] = sf[(row0 + nl) * FEAT + K0];
    a1[ks] = sf[(row0 + nl) * FEAT + K0 + 1];
  }

  // ---- stage 1: (16x64) @ (64x128) + b1, ReLU ----
#pragma unroll 1
  for (int nt = 0; nt < 8; ++nt) {
    __syncthreads();                       // previous tile fully consumed
    for (int e2 = tid; e2 < FEAT * 16; e2 += 128)
      s_b[e2] = w1[(e2 >> 4) * MHID + nt * 16 + (e2 & 15)];
    __syncthreads();
    const int col = nt * 16 + nl;
    const float bias = b1[col];
    v8f acc;
#pragma unroll
    for (int g = 0; g < 8; ++g) acc[g] = bias;
#pragma unroll
    for (int ks = 0; ks < 16; ++ks) {
      const int K0 = ks * 4 + kh * 2;
      v2f a; a.x = a0[ks]; a.y = a1[ks];
      v2f b; b.x = s_b[K0 * 16 + nl]; b.y = s_b[(K0 + 1) * 16 + nl];
      acc = wmma4(a, b, acc);
    }
#pragma unroll
    for (int g = 0; g < 8; ++g) {
      float v = acc[g];
      v = v > 0.f ? v : 0.f;
      s_h[wave][(g + 8 * kh) * MHID + col] = v;
    }
  }

  // ---- stage 2: (16x128) @ (128x256) + b2 ----
#pragma unroll 1
  for (int nt = 0; nt < 16; ++nt) {
    __syncthreads();                       // h ready / previous tile consumed
    for (int e2 = tid; e2 < MHID * 16; e2 += 128)
      s_b[e2] = w2[(e2 >> 4) * MOUT + nt * 16 + (e2 & 15)];
    __syncthreads();
    const int col = nt * 16 + nl;
    const float bias = b2[col];
    v8f acc;
#pragma unroll
    for (int g = 0; g < 8; ++g) acc[g] = bias;
#pragma unroll
    for (int ks = 0; ks < 32; ++ks) {
      const int K0 = ks * 4 + kh * 2;
      v2f a;
      a.x = s_h[wave][nl * MHID + K0];
      a.y = s_h[wave][nl * MHID + K0 + 1];
      v2f b;
      b.x = s_b[K0 * 16 + nl];
      b.y = s_b[(K0 + 1) * 16 + nl];
      acc = wmma4(a, b, acc);
    }
#pragma unroll
    for (int g = 0; g < 8; ++g)
      out[(row0 + g + 8 * kh) * MOUT + col] = acc[g];
  }
}

// ===========================================================================
// Host orchestration
// ===========================================================================
extern "C" void kernel_launch(void* const* d_in, const int* in_sizes, int n_in,
                              void* d_out, int out_size, void* d_ws, size_t ws_size,
                              hipStream_t stream) {
  const float* pts = (const float*)d_in[0];   // 16384 x 3
  const float* ew1 = (const float*)d_in[1];   // 6 x 64
  const float* eb1 = (const float*)d_in[2];   // 64
  const float* ew2 = (const float*)d_in[3];   // 64 x 64
  const float* eb2 = (const float*)d_in[4];   // 64
  const float* mw1 = (const float*)d_in[5];   // 3 x 64 x 128
  const float* mb1 = (const float*)d_in[6];   // 3 x 128
  const float* mw2 = (const float*)d_in[7];   // 3 x 128 x 256
  const float* mb2 = (const float*)d_in[8];   // 3 x 256

  float* out = (float*)d_out;
  float* sp0 = out + 0;           // 4096*3
  float* lf0 = out + 12288;       // 4096*256
  float* sp1 = out + 1060864;     // 1024*3
  float* lf1 = out + 1063936;     // 1024*256
  float* sp2 = out + 1326080;     // 256*3
  float* lf2 = out + 1326848;     // 256*256

  char* ws = (char*)d_ws;
  int*   knn_idx = (int*)(ws + 0);                    // 16384*16 int (1 MB)
  float* feats   = (float*)(ws + (1u << 20));         // 16384*64 f32 (4 MB)
  int*   fi      = (int*)(ws + 5u * (1u << 20));      // up to 4096 int (64 KB)
  float* sf      = (float*)(ws + 5u * (1u << 20) + (1u << 16));  // 4096*64 f32 (1 MB)
  float* ew1p    = (float*)(ws + 6u * (1u << 20) + (1u << 16));  // 8*64 f32 padded

  // 0) zero-pad ew1 -> ew1p (8x64)
  pad_ew1_kernel<<<2, 256, 0, stream>>>(ew1, ew1p);

  // 1) exact KNN (top-16) — one wave / query
  knn_kernel<<<N_PTS / 8, 256, 0, stream>>>(pts, knn_idx);

  // 2) EdgeConv features — one wave / point, WMMA, LDS-resident weights
  edgeconv_kernel<<<N_PTS / 4, 128, 0, stream>>>(pts, knn_idx, ew1p, eb1, ew2, eb2, feats);

  // ---- scale 0 (4096) ----
  fps_kernel<<<1, 1024, 0, stream>>>(pts, N_PTS, 4096, fi);
  gather_rows<<<(4096 * 3 + 255) / 256, 256, 0, stream>>>(pts, fi, sp0, 4096, 3);
  gather_rows<<<(4096 * FEAT + 255) / 256, 256, 0, stream>>>(feats, fi, sf, 4096, FEAT);
  mlp_kernel<<<4096 / 64, 128, 0, stream>>>(sf, mw1 + 0 * 64 * 128, mb1 + 0 * 128,
                                            mw2 + 0 * 128 * 256, mb2 + 0 * 256, lf0);

  // ---- scale 1 (1024), FPS over sp0 ----
  fps_kernel<<<1, 1024, 0, stream>>>(sp0, 4096, 1024, fi);
  gather_rows<<<(1024 * 3 + 255) / 256, 256, 0, stream>>>(sp0, fi, sp1, 1024, 3);
  nn_feats_kernel<<<1024 / 8, 256, 0, stream>>>(sp1, pts, feats, sf);
  mlp_kernel<<<1024 / 64, 128, 0, stream>>>(sf, mw1 + 1 * 64 * 128, mb1 + 1 * 128,
                                            mw2 + 1 * 128 * 256, mb2 + 1 * 256, lf1);

  // ---- scale 2 (256), FPS over sp1 ----
  fps_kernel<<<1, 1024, 0, stream>>>(sp1, 1024, 256, fi);
  gather_rows<<<(256 * 3 + 255) / 256, 256, 0, stream>>>(sp1, fi, sp2, 256, 3);
  nn_feats_kernel<<<256 / 8, 256, 0, stream>>>(sp2, pts, feats, sf);
  mlp_kernel<<<256 / 64, 128, 0, stream>>>(sf, mw1 + 2 * 64 * 128, mb1 + 2 * 128,
                                           mw2 + 2 * 128 * 256, mb2 + 2 * 256, lf2);

  (void)in_sizes; (void)n_in; (void)out_size; (void)ws_size;
}